// CodecModel_3367254360430
// MI455X (gfx1250) — compile-verified
//
#include <hip/hip_runtime.h>
#include <math.h>

#define F_DIM 1024
#define N_DIM 4096
#define KPAT  64
#define NCH   9
#define TBL   (KPAT * NCH)        // 576 entries per table
#define NTBL  5
#define OUT_FLOATS (2 * F_DIM * N_DIM)

// Hardware f32 atomic add into the L2-resident grid (grid = 32MB < 192MB L2).
__device__ __forceinline__ void corner_add(float* __restrict__ out, int fi, int ti,
                                           float w, float vr, float vi) {
  if (fi >= 0 && fi < F_DIM && ti >= 0 && ti < N_DIM) {
    size_t idx = (((size_t)fi * (size_t)N_DIM) + (size_t)ti) * 2u;
    unsafeAtomicAdd(out + idx,     vr * w);   // -> global_atomic_add_f32
    unsafeAtomicAdd(out + idx + 1, vi * w);
  }
}

__global__ void zero_grid(float4* __restrict__ p, int n4) {
  int i = blockIdx.x * blockDim.x + threadIdx.x;
  float4 z; z.x = 0.f; z.y = 0.f; z.z = 0.f; z.w = 0.f;
  if (i < n4) p[i] = z;
}

__global__ __launch_bounds__(256) void deposit_kernel(
    const int*   __restrict__ tau,
    const float* __restrict__ alpha,
    const float* __restrict__ zeta_f,
    const float* __restrict__ zeta_t,
    const float* __restrict__ log_rho,
    const float* __restrict__ theta,
    const float* __restrict__ P_re,
    const float* __restrict__ P_im,
    const float* __restrict__ dF,
    const float* __restrict__ dT,
    const float* __restrict__ logit_gate,
    float* __restrict__ out, int M) {
  // Raw SoA staging area (async-DMA target) + AoS hot tables.
  __shared__ __align__(16) float  s_raw[NTBL * TBL];  // 11.5 KB
  __shared__ __align__(16) float4 s_pk[TBL];          //  9.0 KB: {P_re,P_im,dF,dT}
  __shared__             float  s_gate[TBL];          //  2.3 KB: sigmoid(gate)

  // ---- Stage the pattern bank into LDS with CDNA5 async global->LDS DMA ----
  {
    const float* srcs[NTBL] = {P_re, P_im, dF, dT, logit_gate};
    // Generic pointer to LDS: low 32 bits are the LDS byte offset (ISA 10.2).
    unsigned int lds_base = (unsigned int)(size_t)(&s_raw[0]);
    const int n128 = (NTBL * TBL) / 4;                 // 720 16-byte packets
    for (int t = threadIdx.x; t < n128; t += blockDim.x) {
      int tbl = t / (TBL / 4);
      int e4  = t - tbl * (TBL / 4);
      unsigned long long g = (unsigned long long)(size_t)(srcs[tbl] + e4 * 4);
      unsigned int       l = lds_base + (unsigned int)t * 16u;
      asm volatile("global_load_async_to_lds_b128 %0, %1, off"
                   :: "v"(l), "v"(g) : "memory");      // tracked by ASYNCcnt
    }
    asm volatile("s_wait_asynccnt 0" ::: "memory");    // drain this wave's copies
    __syncthreads();                                   // all waves see staged LDS
    // Repack SoA -> AoS once per block; fold sigmoid(gate) in.
    for (int t = threadIdx.x; t < TBL; t += blockDim.x) {
      float4 e;
      e.x = s_raw[0 * TBL + t];                        // P_re
      e.y = s_raw[1 * TBL + t];                        // P_im
      e.z = s_raw[2 * TBL + t];                        // dF
      e.w = s_raw[3 * TBL + t];                        // dT
      s_pk[t]   = e;
      float x   = s_raw[4 * TBL + t];
      s_gate[t] = 1.0f / (1.0f + __expf(-x));
    }
    __syncthreads();
  }

  int m = blockIdx.x * blockDim.x + threadIdx.x;
  if (m >= M) return;

  // argmax(softmax(alpha/tau)) == argmax(alpha/tau); dividing keeps negative-tau
  // semantics faithful. tau is a 1-element int array (python scalar).
  float inv_tau = 1.0f / (float)tau[0];

  // Each lane owns one contiguous 256B row: 16x b128 loads, all bytes consumed.
  const float4* arow = reinterpret_cast<const float4*>(alpha) + (size_t)m * (KPAT / 4);
  float best = -INFINITY;
  int kbest = 0;
#pragma unroll
  for (int i = 0; i < KPAT / 4; ++i) {
    float4 a = arow[i];
    float v;
    v = a.x * inv_tau; if (v > best) { best = v; kbest = 4 * i + 0; }
    v = a.y * inv_tau; if (v > best) { best = v; kbest = 4 * i + 1; }
    v = a.z * inv_tau; if (v > best) { best = v; kbest = 4 * i + 2; }
    v = a.w * inv_tau; if (v > best) { best = v; kbest = 4 * i + 3; }
  }

  float fc  = zeta_f[m] * 162.97466172610083f;   // F_DIM / (2*pi)
  float nc  = zeta_t[m] * 651.8986469044033f;    // N_DIM / (2*pi)
  float rho = __expf(log_rho[m]);
  float sn, cs;
  __sincosf(theta[m], &sn, &cs);
  float ar = rho * cs, ai = rho * sn;            // amp = rho * e^{i*theta}

  int kb = kbest * NCH;
#pragma unroll
  for (int c = 0; c < NCH; ++c) {
    float4 e = s_pk[kb + c];                     // one ds_load_b128
    float  g = s_gate[kb + c];                   // one ds_load_b32
    float vr = (ar * e.x - ai * e.y) * g;        // amp * P * gate (complex)
    float vi = (ar * e.y + ai * e.x) * g;
    float fh = fc + e.z, nh = nc + e.w;
    float f0 = floorf(fh), t0 = floorf(nh);
    float wf = fh - f0,  wt = nh - t0;
    int f0i = (int)f0, t0i = (int)t0;
    corner_add(out, f0i,     t0i,     (1.f - wf) * (1.f - wt), vr, vi);
    corner_add(out, f0i,     t0i + 1, (1.f - wf) * wt,         vr, vi);
    corner_add(out, f0i + 1, t0i,     wf * (1.f - wt),         vr, vi);
    corner_add(out, f0i + 1, t0i + 1, wf * wt,                 vr, vi);
  }
}

extern "C" void kernel_launch(void* const* d_in, const int* in_sizes, int n_in,
                              void* d_out, int out_size, void* d_ws, size_t ws_size,
                              hipStream_t stream) {
  const int*   tau     = (const int*)  d_in[0];
  const float* alpha   = (const float*)d_in[1];
  const float* zeta_f  = (const float*)d_in[2];
  const float* zeta_t  = (const float*)d_in[3];
  const float* log_rho = (const float*)d_in[4];
  const float* theta   = (const float*)d_in[5];
  const float* P_re    = (const float*)d_in[6];
  const float* P_im    = (const float*)d_in[7];
  const float* dF      = (const float*)d_in[8];
  const float* dT      = (const float*)d_in[9];
  const float* gate    = (const float*)d_in[10];
  float* out = (float*)d_out;
  int M = in_sizes[1] / KPAT;

  // Deterministic per call: re-zero the accumulation grid, then deposit.
  int n4 = OUT_FLOATS / 4;
  zero_grid<<<(n4 + 255) / 256, 256, 0, stream>>>((float4*)out, n4);
  deposit_kernel<<<(M + 255) / 256, 256, 0, stream>>>(
      tau, alpha, zeta_f, zeta_t, log_rho, theta,
      P_re, P_im, dF, dT, gate, out, M);
}